// GraphAttentionLayer_90099823936297
// MI455X (gfx1250) — compile-verified
//
#include <hip/hip_runtime.h>
#include <hip/hip_bf16.h>

// ---------------------------------------------------------------------------
// GAT layer for MI455X (gfx1250, wave32).
//   Wh = h @ W^T + fused alpha projections : f32 WMMA 16x16x4 GEMM
//   edge passes  : leaky-relu, weighted, segment max/sum via atomics
//   scatter      : h_prime[dst] += att * Wh[src]  (atomic adds, L2-resident)
// ---------------------------------------------------------------------------

typedef __attribute__((ext_vector_type(2))) float v2f;
typedef __attribute__((ext_vector_type(8))) float v8f;

#define D_FEAT 128
#define ALPHA_LRELU 0.2f
#define EPS_F 1e-8f

// ---------------------------------------------------------------------------
// Zeroing kernel (harness poisons d_out / d_ws with 0xAA before timing)
// ---------------------------------------------------------------------------
__global__ void zero_f32(float* __restrict__ p, long long n) {
  long long i = (long long)blockIdx.x * blockDim.x + threadIdx.x;
  if (i < n) p[i] = 0.0f;
}

// ---------------------------------------------------------------------------
// Kernel 1: Wh = h @ W^T via V_WMMA_F32_16X16X4_F32, with fused
//   alpha_src[n] = Wh[n]·a[0:128], alpha_dst[n] = Wh[n]·a[128:256].
//
// One block = 8 waves; block b owns row-tile b (16 rows), wave w owns column
// tile w (16 cols) -> the block covers the full 16x128 output stripe, so the
// alpha dot products can be completed block-locally via LDS accumulation.
//
// A frag (16x4):  lane m=lane&15, half=lane>>4 ; elems = h[row0+m][k+2*half+{0,1}]
// B frag (4x16):  K = vgpr + 2*half, N = lane&15 ; B[K][N] = W[col0+N][K]
//                 -> elems = W[col0+m][k+2*half+{0,1}] (same addressing as A)
// C/D (16x16):    vgpr v -> M = v + 8*half, N = lane&15
// ---------------------------------------------------------------------------
__global__ void gemm_wh_wmma(const float* __restrict__ h,
                             const float* __restrict__ W,
                             const float* __restrict__ a,
                             float* __restrict__ Wh,
                             float* __restrict__ alpha_src,
                             float* __restrict__ alpha_dst,
                             int nNodes) {
  const int wave = threadIdx.x >> 5;     // 0..7 -> column tile
  const int lane = threadIdx.x & 31;
  const int m    = lane & 15;
  const int half = lane >> 4;

  const int row0 = blockIdx.x * 16;
  const int col0 = wave * 16;

  __shared__ float s_acc[32];            // [0..15] = src rows, [16..31] = dst rows
  if (threadIdx.x < 32) s_acc[threadIdx.x] = 0.0f;
  __syncthreads();

  // Clamp row for loads (only matters for a remainder tile; stores predicated).
  int rload = row0 + m;
  if (rload >= nNodes) rload = nNodes - 1;

  const float* __restrict__ hrow = h + (long long)rload * D_FEAT;
  const float* __restrict__ wrow = W + (long long)(col0 + m) * D_FEAT;

  v8f c = {};
#pragma unroll 8
  for (int k = 0; k < D_FEAT; k += 4) {
    const int off = k + 2 * half;
    v2f av = *(const v2f*)(hrow + off);  // 8B aligned (off is even)
    v2f bv = *(const v2f*)(wrow + off);
    // 8 args: (neg_a, A, neg_b, B, c_mod, C, reuse_a, reuse_b)
    c = __builtin_amdgcn_wmma_f32_16x16x4_f32(
        false, av, false, bv, (short)0, c, false, false);
  }

  // ---- store Wh tile: fast unguarded path for full tiles (N%16==0 case) ----
  if (row0 + 16 <= nNodes) {
    float* __restrict__ base =
        Wh + (long long)(row0 + 8 * half) * D_FEAT + col0 + m;
#pragma unroll
    for (int v = 0; v < 8; ++v) base[(long long)v * D_FEAT] = c[v];
  } else {
#pragma unroll
    for (int v = 0; v < 8; ++v) {
      const int r = row0 + v + 8 * half;
      if (r < nNodes) Wh[(long long)r * D_FEAT + col0 + m] = c[v];
    }
  }

  // ---- fused alpha projections: reduce c[v]*a[col] over the 16 cols/half ----
  const float asrc = a[col0 + m];
  const float adst = a[D_FEAT + col0 + m];
#pragma unroll
  for (int v = 0; v < 8; ++v) {
    float ps = c[v] * asrc;
    float pd = c[v] * adst;
#pragma unroll
    for (int off = 1; off < 16; off <<= 1) {   // xor stays within each half
      ps += __shfl_xor(ps, off, 32);
      pd += __shfl_xor(pd, off, 32);
    }
    if (m == 0) {                               // one lane per half-wave
      atomicAdd(&s_acc[v + 8 * half], ps);
      atomicAdd(&s_acc[16 + v + 8 * half], pd);
    }
  }
  __syncthreads();
  if (threadIdx.x < 32) {
    const int r = row0 + (threadIdx.x & 15);
    if (r < nNodes) {
      if (threadIdx.x < 16) alpha_src[r] = s_acc[threadIdx.x];
      else                  alpha_dst[r] = s_acc[threadIdx.x];
    }
  }
}

// ---------------------------------------------------------------------------
// Kernel 2: edge logits + segment max.
// seg_max pre-zeroed; int-representation atomicMax gives max(0, max e) exactly
// (positive floats order as ints; negative floats are negative ints < 0).
// This also implements the reference's maximum(seg_max, 0) clamp.
// ---------------------------------------------------------------------------
__global__ void edge_logits(const int* __restrict__ src,
                            const int* __restrict__ dst,
                            const float* __restrict__ ew,
                            const float* __restrict__ alpha_src,
                            const float* __restrict__ alpha_dst,
                            float* __restrict__ ebuf,
                            int* __restrict__ seg_max_i,
                            int nEdges) {
  const int e = blockIdx.x * blockDim.x + threadIdx.x;
  if (e >= nEdges) return;
  const int s = src[e];
  const int d = dst[e];
  float v = alpha_src[s] + alpha_dst[d];
  v = (v >= 0.0f) ? v : ALPHA_LRELU * v;   // leaky relu
  v *= ew[e];
  ebuf[e] = v;
  atomicMax(seg_max_i + d, __float_as_int(v));
}

// ---------------------------------------------------------------------------
// Kernel 3: ex = exp(e - seg_max[dst]); seg_sum[dst] += ex  (ebuf overwritten)
// ---------------------------------------------------------------------------
__global__ void edge_exp_sum(const int* __restrict__ dst,
                             float* __restrict__ ebuf,
                             const float* __restrict__ seg_max,
                             float* __restrict__ seg_sum,
                             int nEdges) {
  const int e = blockIdx.x * blockDim.x + threadIdx.x;
  if (e >= nEdges) return;
  const int d = dst[e];
  const float ex = expf(ebuf[e] - seg_max[d]);
  ebuf[e] = ex;
  atomicAdd(&seg_sum[d], ex);
}

// ---------------------------------------------------------------------------
// Kernel 4: h_prime[dst] += (ex / (seg_sum[dst]+eps)) * Wh[src]
// One wave per edge; float4 load of Wh row; 4 atomic adds per lane.
// Wh + out are L2-resident (51 MB each vs 192 MB L2) -> atomic-throughput bound.
// ---------------------------------------------------------------------------
__global__ void edge_scatter(const float* __restrict__ Wh,
                             const int* __restrict__ src,
                             const int* __restrict__ dst,
                             const float* __restrict__ exbuf,
                             const float* __restrict__ seg_sum,
                             float* __restrict__ out,
                             int nEdges) {
  const int wave = threadIdx.x >> 5;
  const int lane = threadIdx.x & 31;
  const long long e = (long long)blockIdx.x * 8 + wave;
  if (e >= nEdges) return;

  const int s = src[e];
  const int d = dst[e];
  const float att = exbuf[e] / (seg_sum[d] + EPS_F);

  const float4 v = ((const float4*)(Wh + (long long)s * D_FEAT))[lane];
  float* __restrict__ orow = out + (long long)d * D_FEAT + lane * 4;
  atomicAdd(orow + 0, att * v.x);
  atomicAdd(orow + 1, att * v.y);
  atomicAdd(orow + 2, att * v.z);
  atomicAdd(orow + 3, att * v.w);
}

// ---------------------------------------------------------------------------
// Host launcher
// inputs: h[N*128] f32, edge_index[2*E] i32, edge_weight[E] f32,
//         W[128*128] f32, a[256] f32 ; output: h_prime[N*128] f32
// ---------------------------------------------------------------------------
extern "C" void kernel_launch(void* const* d_in, const int* in_sizes, int n_in,
                              void* d_out, int out_size, void* d_ws, size_t ws_size,
                              hipStream_t stream) {
  const float* h  = (const float*)d_in[0];
  const int*   ei = (const int*)d_in[1];
  const float* ew = (const float*)d_in[2];
  const float* W  = (const float*)d_in[3];
  const float* a  = (const float*)d_in[4];
  float* out = (float*)d_out;

  const int N = in_sizes[0] / D_FEAT;
  const int E = in_sizes[1] / 2;
  const int* src = ei;
  const int* dst = ei + E;

  // Workspace carve-up (floats): Wh | alpha_src | alpha_dst | seg_max | seg_sum | ebuf
  float* Wh        = (float*)d_ws;
  float* alpha_src = Wh + (long long)N * D_FEAT;
  float* alpha_dst = alpha_src + N;
  float* seg_max   = alpha_dst + N;
  float* seg_sum   = seg_max + N;
  float* ebuf      = seg_sum + N;

  const long long outElems = (long long)N * D_FEAT;

  // 0) zero output and segment buffers (seg_max/seg_sum are contiguous 2N)
  zero_f32<<<dim3((unsigned)((outElems + 255) / 256)), dim3(256), 0, stream>>>(
      out, outElems);
  zero_f32<<<dim3((unsigned)((2LL * N + 255) / 256)), dim3(256), 0, stream>>>(
      seg_max, 2LL * N);

  // 1) Wh = h @ W^T via WMMA + fused alpha projections
  //    one block per 16-row tile (N=100000 -> 6250 blocks, 8 waves each)
  gemm_wh_wmma<<<dim3((N + 15) / 16), dim3(256), 0, stream>>>(
      h, W, a, Wh, alpha_src, alpha_dst, N);

  // 2) edge logits + segment max
  edge_logits<<<dim3((E + 255) / 256), dim3(256), 0, stream>>>(
      src, dst, ew, alpha_src, alpha_dst, ebuf, (int*)seg_max, E);

  // 3) exp + segment sum
  edge_exp_sum<<<dim3((E + 255) / 256), dim3(256), 0, stream>>>(
      dst, ebuf, seg_max, seg_sum, E);

  // 4) weighted scatter-add into h_prime (wave per edge)
  edge_scatter<<<dim3((E + 7) / 8), dim3(256), 0, stream>>>(
      Wh, src, dst, ebuf, seg_sum, out, E);
}